// ToFInitEncoder_69587060129876
// MI455X (gfx1250) — compile-verified
//
#include <hip/hip_runtime.h>
#include <hip/hip_bf16.h>
#include <math.h>

// ---------------------------------------------------------------------------
// ToF init encoder for MI455X (gfx1250, wave32).
//
//   d_in[0] hist_BZ2  f32 [64,64,2]
//   d_in[1] mask_BZ   bool(1B) [64,64]
//   d_in[2] fr_BZ4    i32 [64,64,4]
//   d_in[3] H         i32 scalar (480)
//   d_in[4] W         i32 scalar (640)
//   d_in[5] w1        f32 [64,2,1,1]
//   d_in[6] b1        f32 [64]
//   d_in[7] w2        f32 [32,64,3,3]   (OIHW -> row-major [oc][576], kk=ic*9+ky*3+kx)
//   d_in[8] b2        f32 [32]
//   d_out             f32 [64,32,8,8]
//
// Workspace: padded activations hpad [B=64][C=64][10][10] f32 = 1.6 MB.
// ---------------------------------------------------------------------------

typedef __attribute__((ext_vector_type(2))) float v2f;
typedef __attribute__((ext_vector_type(8))) float v8f;

#define B_N     64
#define Z_N     64
#define BASE_C  64
#define OUT_C   32
#define CELLS   64            // 8x8 output grid
#define HPAD_B  (BASE_C * 10 * 10)   // 6400 floats per batch

// --------------------------------------------------------------------------
// Kernel 0: zero the padded activation workspace (borders must be 0 for the
// pad-1 3x3 conv; harness poisons d_ws with 0xAA).
// --------------------------------------------------------------------------
__global__ void zero_ws_kernel(float* __restrict__ ws, int n) {
  for (int i = blockIdx.x * blockDim.x + threadIdx.x; i < n;
       i += gridDim.x * blockDim.x)
    ws[i] = 0.0f;
}

// --------------------------------------------------------------------------
// Kernel 1: one block per (batch, output cell).
// Computes the exact area-average of the last-write-wins rectangle scatter
// over this cell's 60x80 pixel block WITHOUT materializing the dense image,
// then log1p -> 1x1 conv (2->64) -> SiLU, written into the padded layout.
// --------------------------------------------------------------------------
__global__ __launch_bounds__(256) void tof_cell_kernel(
    const float* __restrict__ hist,         // [B,Z,2]
    const unsigned char* __restrict__ mask, // [B,Z] bool
    const int* __restrict__ fr,             // [B,Z,4]
    const int* __restrict__ Hs,
    const int* __restrict__ Ws,
    const float* __restrict__ w1,           // [64,2]
    const float* __restrict__ b1,           // [64]
    float* __restrict__ hpad)               // [B,64,10,10]
{
  const int H  = Hs[0], W = Ws[0];
  const int bh = H >> 3, bw = W >> 3;       // 60, 80
  const int blk  = blockIdx.x;
  const int b    = blk >> 6;
  const int cell = blk & 63;
  const int by   = cell >> 3, bx = cell & 7;
  const int tid  = threadIdx.x;

  __shared__ float zm[Z_N], zv[Z_N];
  __shared__ int   zsy[Z_N], zsx[Z_N], zey[Z_N], zex[Z_N];
  __shared__ int   cand[Z_N + 1];           // [0]=count, then z desc
  __shared__ float redm[256], redv[256];
  __shared__ float xmxv[2];

  // Load zone table (clipped boxes, masked values).
  if (tid < Z_N) {
    const int z = tid;
    const float mk = mask[b * Z_N + z] ? 1.0f : 0.0f;
    zm[z] = hist[(b * Z_N + z) * 2 + 0] * mk;
    zv[z] = hist[(b * Z_N + z) * 2 + 1] * mk;
    const int* bo = fr + (size_t)(b * Z_N + z) * 4;
    int sy = max(bo[0], 0), sx = max(bo[1], 0);
    int ey = min(bo[2], H), ex = min(bo[3], W);
    if (mk == 0.0f) ey = sy;                // masked-off zone never writes
    zsy[z] = sy; zsx[z] = sx; zey[z] = ey; zex[z] = ex;
  }
  __syncthreads();

  // Candidate zones intersecting this cell, highest z first (LWW priority).
  if (tid == 0) {
    const int cy0 = by * bh, cy1 = cy0 + bh;
    const int cx0 = bx * bw, cx1 = cx0 + bw;
    int n = 0;
    for (int z = Z_N - 1; z >= 0; --z) {
      if (zey[z] > zsy[z] && zex[z] > zsx[z] &&
          zsy[z] < cy1 && zey[z] > cy0 && zsx[z] < cx1 && zex[z] > cx0)
        cand[1 + n++] = z;
    }
    cand[0] = n;
  }
  __syncthreads();

  // Per-pixel last-write-wins value, accumulated into the block sum.
  const int   ncand = cand[0];
  const int   npx   = bh * bw;              // 4800
  float sm = 0.0f, sv = 0.0f;
  for (int i = tid; i < npx; i += 256) {
    const int y = by * bh + i / bw;
    const int x = bx * bw + i % bw;
    for (int c = 0; c < ncand; ++c) {
      const int z = cand[1 + c];
      if (y >= zsy[z] && y < zey[z] && x >= zsx[z] && x < zex[z]) {
        sm += zm[z]; sv += zv[z];
        break;                              // highest z wins
      }
    }
  }
  redm[tid] = sm; redv[tid] = sv;
  __syncthreads();
  for (int s = 128; s > 0; s >>= 1) {
    if (tid < s) { redm[tid] += redm[tid + s]; redv[tid] += redv[tid + s]; }
    __syncthreads();
  }
  if (tid == 0) {
    const float inv = 1.0f / (float)npx;    // exact area mean (zeros included)
    xmxv[0] = log1pf(fmaxf(redm[0] * inv, 0.0f));
    xmxv[1] = log1pf(fmaxf(redv[0] * inv, 0.0f));
  }
  __syncthreads();

  // Fused 1x1 conv (2->64) + SiLU into padded [10][10] layout.
  if (tid < BASE_C) {
    const int c = tid;
    float h = xmxv[0] * w1[c * 2 + 0] + xmxv[1] * w1[c * 2 + 1] + b1[c];
    h = h / (1.0f + __expf(-h));            // silu
    hpad[((size_t)b * BASE_C + c) * 100 + (by + 1) * 10 + (bx + 1)] = h;
  }
}

// --------------------------------------------------------------------------
// Kernel 2: 3x3 conv 64->32, pad 1, as implicit GEMM on the fp32 tensor op.
// One workgroup (8 waves) per batch; each wave owns one 16x16 D tile of
// Out[32 x 64] = W[32 x 576] x im2col[576 x 64], K-loop of V_WMMA_F32_16X16X4_F32.
//
// Fragment layouts (ISA 7.12.2, fp32 16x16x4):
//   A: lanes 0-15 -> M=lane,    VGPR0/1 = K=4k+0/4k+1
//      lanes16-31 -> M=lane-16, VGPR0/1 = K=4k+2/4k+3
//   B: lanes 0-15 -> N=lane,    VGPR0/1 = K=4k+0/4k+1   (mirrored halves)
//   D: VGPR j -> M = j + (lane>=16 ? 8 : 0), N = lane&15
// --------------------------------------------------------------------------
__global__ __launch_bounds__(256) void conv3x3_wmma_kernel(
    const float* __restrict__ hpad,   // [B,64,10,10]
    const float* __restrict__ w2,     // [32][576], kk = ic*9 + ky*3 + kx
    const float* __restrict__ b2,     // [32]
    float* __restrict__ out)          // [B,32,8,8]
{
  __shared__ float Hlds[HPAD_B];      // 25.6 KB, whole padded batch in LDS

  const int b   = blockIdx.x;
  const int tid = threadIdx.x;
  for (int i = tid; i < HPAD_B; i += 256)
    Hlds[i] = hpad[(size_t)b * HPAD_B + i];
  __syncthreads();

  const int wave = tid >> 5;          // 8 waves
  const int lane = tid & 31;
  const int ot   = wave >> 2;         // 0..1 : oc tile of 16
  const int pt   = wave & 3;          // 0..3 : spatial tile of 16
  const int half = (lane >= 16);
  const int l15  = lane & 15;
  const int oc0  = ot * 16 + l15;     // A row this lane supplies
  const int p    = pt * 16 + l15;     // B col / D col this lane supplies
  const int y    = p >> 3, x = p & 7;
  const int koff = half ? 2 : 0;

  v8f acc = {0.f, 0.f, 0.f, 0.f, 0.f, 0.f, 0.f, 0.f};
  const float* wrow = w2 + (size_t)oc0 * 576;

  for (int k4 = 0; k4 < 576; k4 += 4) {
    const int kk = k4 + koff;
    // A fragment: two contiguous weights (8B-aligned float2 global load).
    const v2f a = *(const v2f*)(wrow + kk);
    // B fragment: im2col gathered from padded LDS tile (no bounds checks).
    const int ic0 = kk / 9,        r0 = kk - ic0 * 9;
    const int ky0 = r0 / 3,        kx0 = r0 - ky0 * 3;
    const int kk1 = kk + 1;
    const int ic1 = kk1 / 9,       r1 = kk1 - ic1 * 9;
    const int ky1 = r1 / 3,        kx1 = r1 - ky1 * 3;
    v2f bb;
    bb.x = Hlds[ic0 * 100 + (y + ky0) * 10 + (x + kx0)];
    bb.y = Hlds[ic1 * 100 + (y + ky1) * 10 + (x + kx1)];
    // D = A(16x4,f32) * B(4x16,f32) + D  -- exact fp32, matches reference.
    acc = __builtin_amdgcn_wmma_f32_16x16x4_f32(
        /*neg_a=*/false, a, /*neg_b=*/false, bb,
        /*c_mod=*/(short)0, acc, /*reuse_a=*/false, /*reuse_b=*/false);
  }

  // Bias + store. acc[j] holds (oc = ot*16 + j + half*8, p).
#pragma unroll
  for (int j = 0; j < 8; ++j) {
    const int ocj = ot * 16 + j + (half ? 8 : 0);
    out[((size_t)b * OUT_C + ocj) * 64 + p] = acc[j] + b2[ocj];
  }
}

// --------------------------------------------------------------------------
extern "C" void kernel_launch(void* const* d_in, const int* in_sizes, int n_in,
                              void* d_out, int out_size, void* d_ws, size_t ws_size,
                              hipStream_t stream) {
  const float*         hist = (const float*)d_in[0];
  const unsigned char* mask = (const unsigned char*)d_in[1];
  const int*           fr   = (const int*)d_in[2];
  const int*           Hs   = (const int*)d_in[3];
  const int*           Ws   = (const int*)d_in[4];
  const float*         w1   = (const float*)d_in[5];
  const float*         b1   = (const float*)d_in[6];
  const float*         w2   = (const float*)d_in[7];
  const float*         b2   = (const float*)d_in[8];
  float* out  = (float*)d_out;
  float* hpad = (float*)d_ws;                 // 64*64*100*4 = 1.6 MB

  const int nz = B_N * HPAD_B;                // 409600 floats
  zero_ws_kernel<<<(nz + 255) / 256, 256, 0, stream>>>(hpad, nz);
  tof_cell_kernel<<<B_N * CELLS, 256, 0, stream>>>(hist, mask, fr, Hs, Ws,
                                                   w1, b1, hpad);
  conv3x3_wmma_kernel<<<B_N, 256, 0, stream>>>(hpad, w2, b2, out);
}